// GraphNCA_57741540327426
// MI455X (gfx1250) — compile-verified
//
#include <hip/hip_runtime.h>

// FP32 WMMA: D(16x16) = A(16x4) * B(4x16) + C.
// A layout: lane<16 holds row=lane, K={k,k+1} in vgpr{0,1}; lane>=16 holds row=lane-16, K={k+2,k+3}.
// B layout (mirror): lane<16 holds col=lane, rows K={k,k+1}; lane>=16 rows K={k+2,k+3}.
// C/D layout: vgpr r -> row r (lanes 0-15) / row r+8 (lanes 16-31), col = lane&15.
typedef float v2f __attribute__((ext_vector_type(2)));
typedef float v8f __attribute__((ext_vector_type(8)));

#define WMMA_F32(a, b, c) \
  __builtin_amdgcn_wmma_f32_16x16x4_f32(false, (a), false, (b), (short)0, (c), false, false)

#define LDS_STRIDE 34  // even -> 8B-aligned GEMM2 A reads (ds_load_b64), conflict-free mod-64 banks

// ---------------- degree / normalization ----------------

__global__ void k_init_deg(float* __restrict__ deg, int n) {
  int i = blockIdx.x * blockDim.x + threadIdx.x;
  if (i < n) deg[i] = 1.0f;  // self-loop contributes 1 to every node's degree
}

__global__ void k_deg(const long long* __restrict__ dst, float* __restrict__ deg, long long E) {
  long long i = (long long)blockIdx.x * blockDim.x + threadIdx.x;
  long long stride = (long long)gridDim.x * blockDim.x;
  for (; i < E; i += stride) {
    unsafeAtomicAdd(&deg[(int)dst[i]], 1.0f);  // global_atomic_add_f32, L2-resident
  }
}

__global__ void k_dinv(const float* __restrict__ deg, float* __restrict__ dinv, int n) {
  int i = blockIdx.x * blockDim.x + threadIdx.x;
  if (i < n) dinv[i] = rsqrtf(deg[i]);  // deg >= 1 always (self-loops), no zero case
}

// ---------------- g = (x @ W_gcn) * dinv[row];  acc initialized to g (self-loop term) ----

__global__ void k_gcn_gemm(const float* __restrict__ x,    // [N,16]
                           const float* __restrict__ Wg,   // [16,48]
                           const float* __restrict__ dinv, // [N]
                           float* __restrict__ g,          // [N,48]
                           float* __restrict__ acc,        // [N,48]
                           int numTiles, int N) {
  int wave = (int)((blockIdx.x * (long long)blockDim.x + threadIdx.x) >> 5);
  if (wave >= numTiles) return;             // whole-wave guard: EXEC stays all-1s for WMMA
  int lane = threadIdx.x & 31;
  int half = lane >> 4;
  int l16  = lane & 15;
  int rowA = wave * 16 + l16;
  int rowAc = rowA < N ? rowA : N - 1;
  const float* xr = x + (size_t)rowAc * 16;

  v8f c0 = {}, c1 = {}, c2 = {};
#pragma unroll
  for (int k = 0; k < 16; k += 4) {
    int kk = k + 2 * half;
    v2f a; a.x = xr[kk]; a.y = xr[kk + 1];  // merges to global_load_b64
    v2f b;
    b.x = Wg[kk * 48 + l16];       b.y = Wg[(kk + 1) * 48 + l16];
    c0 = WMMA_F32(a, b, c0);
    b.x = Wg[kk * 48 + 16 + l16];  b.y = Wg[(kk + 1) * 48 + 16 + l16];
    c1 = WMMA_F32(a, b, c1);
    b.x = Wg[kk * 48 + 32 + l16];  b.y = Wg[(kk + 1) * 48 + 32 + l16];
    c2 = WMMA_F32(a, b, c2);
  }

  bool fullTile = (wave * 16 + 16) <= N;  // uniform per wave -> no per-row branching
  if (fullTile) {
#pragma unroll
    for (int r = 0; r < 8; ++r) {
      int row = wave * 16 + r + half * 8;
      float s = dinv[row];
      size_t base = (size_t)row * 48 + l16;
      float v0 = c0[r] * s, v1 = c1[r] * s, v2 = c2[r] * s;
      g[base]      = v0;  acc[base]      = v0;
      g[base + 16] = v1;  acc[base + 16] = v1;
      g[base + 32] = v2;  acc[base + 32] = v2;
    }
  } else {
#pragma unroll
    for (int r = 0; r < 8; ++r) {
      int row = wave * 16 + r + half * 8;
      if (row < N) {
        float s = dinv[row];
        size_t base = (size_t)row * 48 + l16;
        float v0 = c0[r] * s, v1 = c1[r] * s, v2 = c2[r] * s;
        g[base]      = v0;  acc[base]      = v0;
        g[base + 16] = v1;  acc[base + 16] = v1;
        g[base + 32] = v2;  acc[base + 32] = v2;
      }
    }
  }
}

// ---- edge scatter: acc[dst] += g[src]; 16 lanes/edge, channels interleaved (c, c+16, c+32)
// so each of the three gathers / atomic groups is a contiguous 64B segment across lanes 0-15.

__global__ void k_scatter(const long long* __restrict__ src, const long long* __restrict__ dst,
                          const float* __restrict__ g, float* __restrict__ acc, long long E) {
  long long total = E << 4;  // 16 work-items per edge, 3 strided channels each
  long long i = (long long)blockIdx.x * blockDim.x + threadIdx.x;
  long long stride = (long long)gridDim.x * blockDim.x;
  for (; i < total; i += stride) {
    long long e = i >> 4;
    int c = (int)(i & 15);
    long long s = src[e], d = dst[e];
    const float* gs = g + s * 48 + c;      // gather (HBM-bound, 64B granules)
    float* ad = acc + d * 48 + c;          // scatter: acc is 96MB -> lives in 192MB L2
    unsafeAtomicAdd(ad,      gs[0]);
    unsafeAtomicAdd(ad + 16, gs[16]);
    unsafeAtomicAdd(ad + 32, gs[32]);
  }
}

// ---------------- fused: features = acc*dinv; out = xx + relu(features@W1+b1)@W2+b2 ----

__global__ void k_mlp(const float* __restrict__ acc, const float* __restrict__ dinv,
                      const float* __restrict__ W1,  // [48,32]
                      const float* __restrict__ b1,  // [32]
                      const float* __restrict__ W2,  // [32,16]
                      const float* __restrict__ b2,  // [16]
                      const float* __restrict__ xx,  // [N,16]
                      float* __restrict__ out,       // [N,16]
                      int numTiles, int N) {
  __shared__ float lds[8 * 16 * LDS_STRIDE];  // per-wave 16x32 hidden tile
  int wib  = threadIdx.x >> 5;
  int wave = blockIdx.x * (blockDim.x >> 5) + wib;
  if (wave >= numTiles) return;
  int lane = threadIdx.x & 31;
  int half = lane >> 4;
  int l16  = lane & 15;
  int rowA = wave * 16 + l16;
  int rowAc = rowA < N ? rowA : N - 1;
  float dA = dinv[rowAc];               // fold final dinv[dst] scale into A-operand load
  const float* fr = acc + (size_t)rowAc * 48;
  float* myLds = lds + wib * (16 * LDS_STRIDE);

  // GEMM1: [16x48] x [48x32], bias folded into accumulator init
  v8f h0, h1;
#pragma unroll
  for (int r = 0; r < 8; ++r) { h0[r] = b1[l16]; h1[r] = b1[16 + l16]; }
#pragma unroll
  for (int k = 0; k < 48; k += 4) {
    int kk = k + 2 * half;
    v2f a; a.x = fr[kk] * dA; a.y = fr[kk + 1] * dA;
    v2f b;
    b.x = W1[kk * 32 + l16];       b.y = W1[(kk + 1) * 32 + l16];
    h0 = WMMA_F32(a, b, h0);
    b.x = W1[kk * 32 + 16 + l16];  b.y = W1[(kk + 1) * 32 + 16 + l16];
    h1 = WMMA_F32(a, b, h1);
  }

  // ReLU, stage hidden in LDS to convert D-layout -> A-layout (DS ops in-order per wave)
#pragma unroll
  for (int r = 0; r < 8; ++r) {
    int row = r + half * 8;
    myLds[row * LDS_STRIDE + l16]      = fmaxf(h0[r], 0.0f);
    myLds[row * LDS_STRIDE + 16 + l16] = fmaxf(h1[r], 0.0f);
  }
  __builtin_amdgcn_wave_barrier();  // keep LDS stores scheduled before the reads

  // GEMM2: [16x32] x [32x16], bias folded into accumulator init
  v8f o;
#pragma unroll
  for (int r = 0; r < 8; ++r) o[r] = b2[l16];
#pragma unroll
  for (int k = 0; k < 32; k += 4) {
    int kk = k + 2 * half;
    v2f a;  // 8B-aligned pair -> ds_load_b64
    a.x = myLds[l16 * LDS_STRIDE + kk]; a.y = myLds[l16 * LDS_STRIDE + kk + 1];
    v2f b; b.x = W2[kk * 16 + l16];     b.y = W2[(kk + 1) * 16 + l16];
    o = WMMA_F32(a, b, o);
  }

  // residual add + store (lanes 0-15 / 16-31 write contiguous 64B row segments)
  bool fullTile = (wave * 16 + 16) <= N;  // uniform per wave -> straight-line epilogue
  if (fullTile) {
#pragma unroll
    for (int r = 0; r < 8; ++r) {
      int row = wave * 16 + r + half * 8;
      size_t idx = (size_t)row * 16 + l16;
      out[idx] = xx[idx] + o[r];
    }
  } else {
#pragma unroll
    for (int r = 0; r < 8; ++r) {
      int row = wave * 16 + r + half * 8;
      if (row < N) {
        size_t idx = (size_t)row * 16 + l16;
        out[idx] = xx[idx] + o[r];
      }
    }
  }
}

// ---------------- launch ----------------

extern "C" void kernel_launch(void* const* d_in, const int* in_sizes, int n_in,
                              void* d_out, int out_size, void* d_ws, size_t ws_size,
                              hipStream_t stream) {
  const float*     xx = (const float*)d_in[0];
  const long long* ei = (const long long*)d_in[1];  // int64 [2,E] flattened
  const float*     Wg = (const float*)d_in[2];
  const float*     W1 = (const float*)d_in[3];
  const float*     b1 = (const float*)d_in[4];
  const float*     W2 = (const float*)d_in[5];
  const float*     b2 = (const float*)d_in[6];
  float* out = (float*)d_out;

  int       N = in_sizes[0] / 16;
  long long E = (long long)in_sizes[1] / 2;
  const long long* src = ei;
  const long long* dst = ei + E;

  // workspace layout (floats): deg[N] | dinv[N] | g[N*48] | acc[N*48]  (~196 MB)
  float* ws   = (float*)d_ws;
  float* deg  = ws;
  float* dinv = ws + N;
  float* g    = ws + 2LL * N;
  float* acc  = g + 48LL * N;

  int numTiles = (N + 15) / 16;  // N=500000 -> 31250, exact

  k_init_deg<<<(N + 255) / 256, 256, 0, stream>>>(deg, N);
  k_deg<<<4096, 256, 0, stream>>>(dst, deg, E);
  k_dinv<<<(N + 255) / 256, 256, 0, stream>>>(deg, dinv, N);
  k_gcn_gemm<<<(numTiles + 7) / 8, 256, 0, stream>>>(xx, Wg, dinv, g, acc, numTiles, N);
  k_scatter<<<65536, 256, 0, stream>>>(src, dst, g, acc, E);
  k_mlp<<<(numTiles + 7) / 8, 256, 0, stream>>>(acc, dinv, W1, b1, W2, b2, xx, out, numTiles, N);
}